// GATEncoder_32925219291500
// MI455X (gfx1250) — compile-verified
//
#include <hip/hip_runtime.h>
#include <hip/hip_bf16.h>
#include <math.h>

typedef __attribute__((ext_vector_type(2))) float v2f;
typedef __attribute__((ext_vector_type(8))) float v8f;

#define NEG_SLOPE 0.2f

// ---------------------------------------------------------------------------
// utility: fill a float buffer with a constant
// ---------------------------------------------------------------------------
__global__ void fill_f32(float* __restrict__ p, float v, long n) {
  long i = (long)blockIdx.x * blockDim.x + threadIdx.x;
  if (i < n) p[i] = v;
}

// ---------------------------------------------------------------------------
// C[M,Ncols] = A[M,K] @ B[K,Ncols], fp32 via V_WMMA_F32_16X16X4_F32.
// One wave per 16x16 output tile. blockDim.x = 32*(Ncols/16), grid.x = M/16.
// M must be a multiple of 16 (100000 = 6250*16) so EXEC is all-ones (WMMA req).
//
// VGPR layouts per CDNA5 ISA 7.12.2:
//   A (16x4 f32): lanes 0-15 hold M=lane, {K=k,K=k+1}; lanes 16-31 hold
//                 M=lane-16, {K=k+2,K=k+3}
//   B (4x16 f32): vreg0 lanes 0-15 = row K=k, lanes 16-31 = row K=k+2;
//                 vreg1 = rows k+1 / k+3; N = lane&15
//   D (16x16 f32): vreg r: lanes 0-15 -> M=r, lanes 16-31 -> M=r+8, N=lane&15
// ---------------------------------------------------------------------------
__global__ void wmma_gemm_f32(const float* __restrict__ A,
                              const float* __restrict__ B,
                              float* __restrict__ C, int K, int Ncols) {
  const int tileM = blockIdx.x * 16;
  const int wave  = threadIdx.x >> 5;
  const int lane  = threadIdx.x & 31;
  const int tileN = wave * 16;
  const int r  = lane & 15;   // row within A tile / col within B,D tile
  const int hi = lane >> 4;   // which K pair / which M half

  const float* __restrict__ Arow = A + (size_t)(tileM + r) * K;
  v8f acc = {};

#pragma unroll 4
  for (int k = 0; k < K; k += 4) {
    v2f a, b;
    const int ka = k + 2 * hi;
    a.x = Arow[ka];
    a.y = Arow[ka + 1];
    b.x = B[(size_t)ka * Ncols + tileN + r];
    b.y = B[(size_t)(ka + 1) * Ncols + tileN + r];
    acc = __builtin_amdgcn_wmma_f32_16x16x4_f32(
        /*neg_a=*/false, a, /*neg_b=*/false, b,
        /*c_mod=*/(short)0, acc, /*reuse_a=*/false, /*reuse_b=*/false);
  }

  const int col     = tileN + r;
  const int rowBase = tileM + (hi << 3);
#pragma unroll
  for (int i = 0; i < 8; ++i)
    C[(size_t)(rowBase + i) * Ncols + col] = acc[i];
}

// ---------------------------------------------------------------------------
// Per-node attention logits: a_src[n,h] = sum_c h[n,h,c]*att_src[h,c] (C=64).
// One wave per (n,h); lanes cover c and c+32; shuffle tree reduction (wave32).
// ---------------------------------------------------------------------------
__global__ void att_logits(const float* __restrict__ h,
                           const float* __restrict__ att_s,
                           const float* __restrict__ att_d,
                           float* __restrict__ a_s, float* __restrict__ a_d,
                           int N, int H) {
  const int wid  = (blockIdx.x * blockDim.x + threadIdx.x) >> 5;
  const int lane = threadIdx.x & 31;
  if (wid >= N * H) return;
  const int n  = wid / H;
  const int hh = wid - n * H;
  const float* __restrict__ hp = h + ((size_t)n * H + hh) * 64;
  const float* __restrict__ ws = att_s + hh * 64;
  const float* __restrict__ wd = att_d + hh * 64;
  float s = hp[lane] * ws[lane] + hp[lane + 32] * ws[lane + 32];
  float d = hp[lane] * wd[lane] + hp[lane + 32] * wd[lane + 32];
#pragma unroll
  for (int off = 16; off > 0; off >>= 1) {
    s += __shfl_down(s, off, 32);
    d += __shfl_down(d, off, 32);
  }
  if (lane == 0) {
    a_s[wid] = s;
    a_d[wid] = d;
  }
}

// sign-split float atomic max (maps to global_atomic_max_i32 / min_u32)
__device__ __forceinline__ void atomicMaxF32(float* addr, float v) {
  if (v >= 0.0f)
    atomicMax((int*)addr, __float_as_int(v));
  else
    atomicMin((unsigned int*)addr, __float_as_uint(v));
}

// ---------------------------------------------------------------------------
// Pass 1: e = leaky_relu(a_src[src]+a_dst[dst]); running max per (dst,head).
// Edges e < E come from edge_index; e >= E are self-loops (src=dst=e-E).
// ---------------------------------------------------------------------------
__global__ void edge_logit_max(const int* __restrict__ ei, int E, int N, int H,
                               const float* __restrict__ a_s,
                               const float* __restrict__ a_d,
                               float* __restrict__ elog,
                               float* __restrict__ mx) {
  const long idx = (long)blockIdx.x * blockDim.x + threadIdx.x;
  const long total = ((long)E + N) * H;
  if (idx >= total) return;
  const long e  = idx / H;
  const int  hh = (int)(idx - e * H);
  const int  s  = (e < E) ? ei[e] : (int)(e - E);
  const int  d  = (e < E) ? ei[(size_t)E + e] : (int)(e - E);
  float v = a_s[(size_t)s * H + hh] + a_d[(size_t)d * H + hh];
  v = (v > 0.0f) ? v : NEG_SLOPE * v;
  elog[idx] = v;
  atomicMaxF32(&mx[(size_t)d * H + hh], v);
}

// ---------------------------------------------------------------------------
// Pass 2: elog <- exp(elog - max[dst]); denom[dst] += elog  (softmax partition)
// ---------------------------------------------------------------------------
__global__ void edge_exp_sum(const int* __restrict__ ei, int E, int N, int H,
                             float* __restrict__ elog,
                             const float* __restrict__ mx,
                             float* __restrict__ dn) {
  const long idx = (long)blockIdx.x * blockDim.x + threadIdx.x;
  const long total = ((long)E + N) * H;
  if (idx >= total) return;
  const long e  = idx / H;
  const int  hh = (int)(idx - e * H);
  const int  d  = (e < E) ? ei[(size_t)E + e] : (int)(e - E);
  const float ex = __expf(elog[idx] - mx[(size_t)d * H + hh]);
  elog[idx] = ex;
  atomicAdd(&dn[(size_t)d * H + hh], ex);
}

// ---------------------------------------------------------------------------
// Pass 3: out[dst,h,:] += (elog/denom[dst]) * hfeat[src,h,:]   (C=64)
// One wave per (edge,head); lanes cover channels c and c+32.
// ---------------------------------------------------------------------------
__global__ void edge_aggregate(const int* __restrict__ ei, int E, int N, int H,
                               const float* __restrict__ elog,
                               const float* __restrict__ dn,
                               const float* __restrict__ hfeat,
                               float* __restrict__ out) {
  const long wid  = ((long)blockIdx.x * blockDim.x + threadIdx.x) >> 5;
  const int  lane = threadIdx.x & 31;
  const long total = ((long)E + N) * H;
  if (wid >= total) return;
  const long e  = wid / H;
  const int  hh = (int)(wid - e * H);
  const int  s  = (e < E) ? ei[e] : (int)(e - E);
  const int  d  = (e < E) ? ei[(size_t)E + e] : (int)(e - E);
  const float alpha = elog[wid] / dn[(size_t)d * H + hh];
  const float* __restrict__ hp = hfeat + ((size_t)s * H + hh) * 64;
  float* __restrict__ op = out + ((size_t)d * H + hh) * 64;
  atomicAdd(&op[lane],      alpha * hp[lane]);
  atomicAdd(&op[lane + 32], alpha * hp[lane + 32]);
}

// out[i] = elu(out[i] + b[i % F])
__global__ void bias_elu(float* __restrict__ x, const float* __restrict__ b,
                         long n, int F) {
  long i = (long)blockIdx.x * blockDim.x + threadIdx.x;
  if (i >= n) return;
  float v = x[i] + b[i % F];
  x[i] = (v > 0.0f) ? v : expm1f(v);
}

// out[i] += b[i % F]
__global__ void bias_add(float* __restrict__ x, const float* __restrict__ b,
                         long n, int F) {
  long i = (long)blockIdx.x * blockDim.x + threadIdx.x;
  if (i >= n) return;
  x[i] = x[i] + b[i % F];
}

static inline int cdiv_l(long a, int b) { return (int)((a + b - 1) / b); }

extern "C" void kernel_launch(void* const* d_in, const int* in_sizes, int n_in,
                              void* d_out, int out_size, void* d_ws, size_t ws_size,
                              hipStream_t stream) {
  const float* x    = (const float*)d_in[0];
  const int*   ei   = (const int*)d_in[1];
  const float* W1   = (const float*)d_in[2];
  const float* as1w = (const float*)d_in[3];
  const float* ad1w = (const float*)d_in[4];
  const float* b1   = (const float*)d_in[5];
  const float* W2   = (const float*)d_in[6];
  const float* as2w = (const float*)d_in[7];
  const float* ad2w = (const float*)d_in[8];
  const float* b2   = (const float*)d_in[9];

  const int Fin = 128;
  const int N   = in_sizes[0] / Fin;  // 100000 (multiple of 16)
  const int E   = in_sizes[1] / 2;    // 1600000
  const long Et = (long)E + N;        // edges + self loops

  // workspace layout (floats); softmax scratch reused between layers
  float* ws   = (float*)d_ws;
  float* h1   = ws;  ws += (size_t)N * 128;  // x @ W1            [N,2,64]
  float* agg1 = ws;  ws += (size_t)N * 128;  // layer-1 output    [N,128]
  float* h2   = ws;  ws += (size_t)N * 64;   // agg1 @ W2         [N,1,64]
  float* a_s  = ws;  ws += (size_t)N * 2;    // per-node src logits
  float* a_d  = ws;  ws += (size_t)N * 2;    // per-node dst logits
  float* mx   = ws;  ws += (size_t)N * 2;    // segment max
  float* dn   = ws;  ws += (size_t)N * 2;    // segment sum
  float* el   = ws;  ws += (size_t)Et * 2;   // per-edge logits / exp
  float* outp = (float*)d_out;

  const float NEG_INF = -3.402823466e+38f;

  // ----------------------- layer 1 (H=2, C=64) ---------------------------
  {
    const int H = 2;
    wmma_gemm_f32<<<N / 16, 32 * (128 / 16), 0, stream>>>(x, W1, h1, 128, 128);
    att_logits<<<cdiv_l((long)N * H, 8), 256, 0, stream>>>(h1, as1w, ad1w,
                                                           a_s, a_d, N, H);
    fill_f32<<<cdiv_l((long)N * H, 256), 256, 0, stream>>>(mx, NEG_INF, (long)N * H);
    fill_f32<<<cdiv_l((long)N * H, 256), 256, 0, stream>>>(dn, 0.0f, (long)N * H);
    fill_f32<<<cdiv_l((long)N * 128, 256), 256, 0, stream>>>(agg1, 0.0f, (long)N * 128);
    edge_logit_max<<<cdiv_l(Et * H, 256), 256, 0, stream>>>(ei, E, N, H,
                                                            a_s, a_d, el, mx);
    edge_exp_sum<<<cdiv_l(Et * H, 256), 256, 0, stream>>>(ei, E, N, H, el, mx, dn);
    edge_aggregate<<<cdiv_l(Et * H * 32, 256), 256, 0, stream>>>(ei, E, N, H,
                                                                 el, dn, h1, agg1);
    bias_elu<<<cdiv_l((long)N * 128, 256), 256, 0, stream>>>(agg1, b1,
                                                             (long)N * 128, 128);
  }

  // ----------------------- layer 2 (H=1, C=64) ---------------------------
  {
    const int H = 1;
    wmma_gemm_f32<<<N / 16, 32 * (64 / 16), 0, stream>>>(agg1, W2, h2, 128, 64);
    att_logits<<<cdiv_l((long)N * H, 8), 256, 0, stream>>>(h2, as2w, ad2w,
                                                           a_s, a_d, N, H);
    fill_f32<<<cdiv_l((long)N * H, 256), 256, 0, stream>>>(mx, NEG_INF, (long)N * H);
    fill_f32<<<cdiv_l((long)N * H, 256), 256, 0, stream>>>(dn, 0.0f, (long)N * H);
    fill_f32<<<cdiv_l((long)N * 64, 256), 256, 0, stream>>>(outp, 0.0f, (long)N * 64);
    edge_logit_max<<<cdiv_l(Et * H, 256), 256, 0, stream>>>(ei, E, N, H,
                                                            a_s, a_d, el, mx);
    edge_exp_sum<<<cdiv_l(Et * H, 256), 256, 0, stream>>>(ei, E, N, H, el, mx, dn);
    edge_aggregate<<<cdiv_l(Et * H * 32, 256), 256, 0, stream>>>(ei, E, N, H,
                                                                 el, dn, h2, outp);
    bias_add<<<cdiv_l((long)N * 64, 256), 256, 0, stream>>>(outp, b2,
                                                            (long)N * 64, 64);
  }
}